// MeshNN_29042568855825
// MI455X (gfx1250) — compile-verified
//
#include <hip/hip_runtime.h>

typedef float v2f __attribute__((ext_vector_type(2)));
typedef float v8f __attribute__((ext_vector_type(8)));

#define TPc (1.0f - 1.0f / 150.0f)
#define TMc (1.0f + 1.0f / 150.0f)

__device__ __forceinline__ float relu_f(float v) { return fmaxf(v, 0.0f); }

// hat evaluated from precomputed (xi, invLeft=-1/(xi-xm), invRight=-1/(xp-xi))
__device__ __forceinline__ float hat_pre(float x, float xi, float il, float ir) {
    float l = relu_f(relu_f(xi - x) * il + 1.0f);
    float r = relu_f(relu_f(x - xi) * ir + 1.0f);
    return l + r - 1.0f;
}

__global__ __launch_bounds__(256) void meshnn_sparse_wmma(
    const float* __restrict__ xg,      // [n] points
    const float* __restrict__ coords,  // [258] nodes
    const float* __restrict__ wuu,     // [256] interior weights
    const float* __restrict__ wdd,     // [2]   boundary weights
    float* __restrict__ out,           // [n]
    int n)
{
    __shared__ float sc[258];                     // node coordinates
    __shared__ float sxi[256], sil[256], sir[256]; // clamped centers + slopes
    __shared__ float sw[256];                      // interior weights
    __shared__ float sb[8];                        // boundary consts: xi0,il0,ir0,xiL,ilL,irL,w0,wL

    const int t = threadIdx.x;

    // ---- stage node table (258 floats) ----
    sc[t] = coords[t];
    if (t < 2) sc[256 + t] = coords[256 + t];
    __syncthreads();

    // ---- precompute per-hat clamped center & reciprocal slopes ----
    {
        float xm = sc[t];
        float xc = sc[t + 1];
        float xp = sc[t + 2];
        float xi = fmaxf(fminf(xc, TPc * xp), TMc * xm);
        sxi[t] = xi;
        sil[t] = -1.0f / (xi - xm);
        sir[t] = -1.0f / (xp - xi);
        sw[t]  = wuu[t];
    }
    if (t == 0) {
        float c0 = sc[0], c1 = sc[1], cL = sc[257], cP = sc[256];
        float xm0 = c0 - cL * 0.01f;                       // c[0] - c[-1]/100
        float xi0 = fmaxf(fminf(c0, TPc * c1), TMc * xm0);
        sb[0] = xi0;
        sb[1] = -1.0f / (xi0 - xm0);
        sb[2] = -1.0f / (c1 - xi0);
        float xpL = cL * 1.01f;                            // c[-1]*(1+1/100)
        float xiL = fmaxf(fminf(cL, TPc * xpL), TMc * cP);
        sb[3] = xiL;
        sb[4] = -1.0f / (xiL - cP);
        sb[5] = -1.0f / (xpL - xiL);
        sb[6] = wdd[0];
        sb[7] = wdd[1];
    }
    __syncthreads();

    // ---- one 16-point tile per wave32 ----
    const int gtid = blockIdx.x * blockDim.x + t;
    const int wave = gtid >> 5;
    const int lane = t & 31;
    const int p    = lane & 15;          // point within tile
    const bool lo  = lane < 16;          // low half: interior hats; high half: boundary hats

    const int base = wave * 16;
    const int idx  = base + p;
    const float x  = xg[min(idx, n - 1)];

    // interval index j in [0,256]: guess then correct against LDS node table
    int j = (int)(x * (257.0f / 10.0f));
    j = min(max(j, 0), 256);
    if (j > 0   && x <  sc[j])     --j;
    if (j > 0   && x <  sc[j])     --j;
    if (j < 256 && x >= sc[j + 1]) ++j;
    if (j < 256 && x >= sc[j + 1]) ++j;

    // interior candidates: i = j-1 and i = j (weight-masked when out of range)
    const int iA = max(j - 1, 0);
    const int iB = min(j, 255);
    const float phiA = hat_pre(x, sxi[iA], sil[iA], sir[iA]);
    const float phiB = hat_pre(x, sxi[iB], sil[iB], sir[iB]);
    const float wA = (j >= 1)   ? sw[iA] : 0.0f;
    const float wB = (j <= 255) ? sw[iB] : 0.0f;

    // boundary candidates (support-limited: evaluate ~0 away from the ends)
    const float phi0 = hat_pre(x, sb[0], sb[1], sb[2]);
    const float phiL = hat_pre(x, sb[3], sb[4], sb[5]);

    // A (16x4 f32): M = lane&15; lanes 0-15 hold K=0,1 ; lanes 16-31 hold K=2,3
    // B (4x16 f32): N = lane&15; same half-wave K split -> lane-local candidate weights
    v2f A, B;
    A.x = lo ? phiA : phi0;
    A.y = lo ? phiB : phiL;
    B.x = lo ? wA   : sb[6];
    B.y = lo ? wB   : sb[7];

    v8f acc = {};
    // D[m][n] = sum_k phi[m][k] * w_cand(n)[k]  -> diagonal is u[m]; exact fp32 FMA
    acc = __builtin_amdgcn_wmma_f32_16x16x4_f32(
        false, A, false, B, (short)0, acc, false, false);

    // diagonal extraction: m<8 lives at (VGPR m, lane m); m>=8 at (VGPR m-8, lane m+16)
    const bool active = (lane < 8) || (lane >= 24);
    const int  sel    = (lane < 8) ? lane : ((lane >= 24) ? (lane - 24) : 0);
    const int  m      = (lane < 8) ? lane : (lane - 16);

    float u = acc[0];
#pragma unroll
    for (int v = 1; v < 8; ++v) u = (sel == v) ? acc[v] : u;

    const int oidx = base + m;
    if (active && oidx < n) out[oidx] = u;
}

extern "C" void kernel_launch(void* const* d_in, const int* in_sizes, int n_in,
                              void* d_out, int out_size, void* d_ws, size_t ws_size,
                              hipStream_t stream) {
    (void)n_in; (void)d_ws; (void)ws_size; (void)out_size;
    const float* x      = (const float*)d_in[0];
    const float* coords = (const float*)d_in[1];
    const float* wuu    = (const float*)d_in[2];
    const float* wdd    = (const float*)d_in[3];
    float* out = (float*)d_out;

    const int n      = in_sizes[0];          // 500000 points
    const int waves  = (n + 15) / 16;        // one 16-pt tile per wave32
    const int blocks = (waves + 7) / 8;      // 256 threads = 8 waves per block
    meshnn_sparse_wmma<<<blocks, 256, 0, stream>>>(x, coords, wuu, wdd, out, n);
}